// Phi_14723147891163
// MI455X (gfx1250) — compile-verified
//
#include <hip/hip_runtime.h>

// Problem constants (from setup_inputs)
#define NEX  4096
#define DD   63
#define DIN  64
#define MM   256
#define RR   10
#define HH   0.5f
#define NB   16      // samples per workgroup in batched WMMA GEMMs

typedef __attribute__((ext_vector_type(2))) float v2f;
typedef __attribute__((ext_vector_type(8))) float v8f;

__device__ __forceinline__ float adtanh(float x) {
    float a = fabsf(x);
    return a + log1pf(expf(-2.0f * a));
}

// ---------------------------------------------------------------- setup ----
// symA = A^T A (64x64), rowsum2[m] = sum_{k<63} W0[m][k]^2, traceA = tr(symA[:63,:63])
__global__ void setup_kernel(const float* __restrict__ A, const float* __restrict__ W0,
                             float* __restrict__ symA, float* __restrict__ rowsum2,
                             float* __restrict__ traceA) {
    int tid = threadIdx.x;
    float rs = 0.f;
    for (int dd = 0; dd < DD; ++dd) { float w = W0[tid * DIN + dd]; rs += w * w; }
    rowsum2[tid] = rs;
    for (int idx = tid; idx < DIN * DIN; idx += 256) {
        int j = idx >> 6, k = idx & 63;
        float s = 0.f;
        for (int r = 0; r < RR; ++r) s += A[r * DIN + j] * A[r * DIN + k];
        symA[idx] = s;
    }
    if (tid == 0) {
        float tr = 0.f;
        for (int k = 0; k < DD; ++k)
            for (int r = 0; r < RR; ++r) { float a = A[r * DIN + k]; tr += a * a; }
        traceA[0] = tr;
    }
}

// ------------------------------------------------------------------- xf ----
__global__ void xf_kernel(const float* __restrict__ x, const float* __restrict__ t,
                          float* __restrict__ xf) {
    int i = blockIdx.x * blockDim.x + threadIdx.x;
    if (i >= NEX * DIN) return;
    int n = i >> 6, k = i & 63;
    xf[i] = (k < DD) ? x[n * DD + k] : t[0];
}

// ----------------------------------------------- opening (WMMA, batched) ----
// pre(16xMM) = xf_tile(16xDIN) @ W0^T ; tanhop = tanh(pre+b0) ; u0 = adtanh(pre+b0)
__global__ __launch_bounds__(256) void opening_wmma_kernel(
    const float* __restrict__ xf, const float* __restrict__ W0,
    const float* __restrict__ bias, float* __restrict__ tanhop, float* __restrict__ u0) {
    __shared__ float xs[NB * DIN];                    // 4 KB
    const int n0 = blockIdx.x * NB;
    const int tid = threadIdx.x, lane = tid & 31, wave = tid >> 5;
    const int laneLo = lane & 15, koff = (lane >> 4) << 1, mhalf = (lane >> 4) << 3;

    for (int i = tid; i < NB * DIN; i += 256) xs[i] = xf[n0 * DIN + i];
    __syncthreads();

    const v8f vz = {0, 0, 0, 0, 0, 0, 0, 0};
    v8f d0 = vz, d1 = vz;
    const int c0 = (wave * 2 + 0) * 16 + laneLo;      // output column (m)
    const int c1 = (wave * 2 + 1) * 16 + laneLo;
    for (int k0 = 0; k0 < DIN; k0 += 4) {
        v2f a   = *reinterpret_cast<const v2f*>(&xs[laneLo * DIN + k0 + koff]);
        v2f bf0 = *reinterpret_cast<const v2f*>(W0 + c0 * DIN + k0 + koff);   // B[k][m]=W0[m][k]
        v2f bf1 = *reinterpret_cast<const v2f*>(W0 + c1 * DIN + k0 + koff);
        d0 = __builtin_amdgcn_wmma_f32_16x16x4_f32(false, a, false, bf0, (short)0, d0, false, false);
        d1 = __builtin_amdgcn_wmma_f32_16x16x4_f32(false, a, false, bf1, (short)0, d1, false, false);
    }
    float bb0 = bias[c0], bb1 = bias[c1];
    #pragma unroll
    for (int v = 0; v < 8; ++v) {
        int smp = mhalf + v;
        float p0 = d0[v] + bb0, p1 = d1[v] + bb1;
        int i0 = (n0 + smp) * MM + c0, i1 = (n0 + smp) * MM + c1;
        tanhop[i0] = tanhf(p0);  u0[i0] = adtanh(p0);
        tanhop[i1] = tanhf(p1);  u0[i1] = adtanh(p1);
    }
}

// ------------------------------------------ resnet layer (WMMA, batched) ----
// pre = u_tile(16xMM) @ Wl^T + bl ; tp_out = tanh(pre) ; u_out = u + h*adtanh(pre)
__global__ __launch_bounds__(256) void layer_wmma_kernel(
    const float* __restrict__ uin, const float* __restrict__ Wl,
    const float* __restrict__ bl, float* __restrict__ tp_out, float* u_out) {
    __shared__ float us[NB * MM];                     // 16 KB
    const int n0 = blockIdx.x * NB;
    const int tid = threadIdx.x, lane = tid & 31, wave = tid >> 5;
    const int laneLo = lane & 15, koff = (lane >> 4) << 1, mhalf = (lane >> 4) << 3;

    for (int i = tid; i < NB * MM; i += 256) us[i] = uin[n0 * MM + i];
    __syncthreads();

    const v8f vz = {0, 0, 0, 0, 0, 0, 0, 0};
    v8f d0 = vz, d1 = vz;
    const int c0 = (wave * 2 + 0) * 16 + laneLo;
    const int c1 = (wave * 2 + 1) * 16 + laneLo;
    for (int k0 = 0; k0 < MM; k0 += 4) {
        v2f a   = *reinterpret_cast<const v2f*>(&us[laneLo * MM + k0 + koff]);
        v2f bf0 = *reinterpret_cast<const v2f*>(Wl + c0 * MM + k0 + koff);    // B[k][m]=Wl[m][k]
        v2f bf1 = *reinterpret_cast<const v2f*>(Wl + c1 * MM + k0 + koff);
        d0 = __builtin_amdgcn_wmma_f32_16x16x4_f32(false, a, false, bf0, (short)0, d0, false, false);
        d1 = __builtin_amdgcn_wmma_f32_16x16x4_f32(false, a, false, bf1, (short)0, d1, false, false);
    }
    float bb0 = bl[c0], bb1 = bl[c1];
    #pragma unroll
    for (int v = 0; v < 8; ++v) {
        int smp = mhalf + v;
        float p0 = d0[v] + bb0, p1 = d1[v] + bb1;
        int i0 = (n0 + smp) * MM + c0, i1 = (n0 + smp) * MM + c1;
        tp_out[i0] = tanhf(p0);
        tp_out[i1] = tanhf(p1);
        if (u_out) {
            u_out[i0] = us[smp * MM + c0] + HH * adtanh(p0);
            u_out[i1] = us[smp * MM + c1] + HH * adtanh(p1);
        }
    }
}

// -------------------------------------------- z backward (WMMA, batched) ----
// zout[n][a] = zv[n][a] + h * sum_m (tp[n][m]*zv[n][m]) * Wl[m][a]
// zin == nullptr -> zv = w_w broadcast (first backward step).
__global__ __launch_bounds__(256) void zback_wmma_kernel(
    const float* __restrict__ tpin, const float* zin, const float* __restrict__ w_w,
    const float* __restrict__ Wl, float* __restrict__ zout) {
    __shared__ float ss[NB * MM];                     // 16 KB
    const int n0 = blockIdx.x * NB;
    const int tid = threadIdx.x, lane = tid & 31, wave = tid >> 5;
    const int laneLo = lane & 15, koff = (lane >> 4) << 1, mhalf = (lane >> 4) << 3;

    for (int i = tid; i < NB * MM; i += 256) {
        int m = i & (MM - 1);
        float zv = zin ? zin[n0 * MM + i] : w_w[m];
        ss[i] = tpin[n0 * MM + i] * zv;
    }
    __syncthreads();

    const v8f vz = {0, 0, 0, 0, 0, 0, 0, 0};
    v8f d0 = vz, d1 = vz;
    const int a0 = (wave * 2 + 0) * 16 + laneLo;      // output column (a)
    const int a1 = (wave * 2 + 1) * 16 + laneLo;
    for (int k0 = 0; k0 < MM; k0 += 4) {
        v2f af = *reinterpret_cast<const v2f*>(&ss[laneLo * MM + k0 + koff]);
        // NN GEMM: B[k][a] = Wl[k*MM+a]; K-adjacent elements are MM apart -> two b32 loads
        v2f bf0, bf1;
        bf0[0] = Wl[(k0 + koff)     * MM + a0];
        bf0[1] = Wl[(k0 + koff + 1) * MM + a0];
        bf1[0] = Wl[(k0 + koff)     * MM + a1];
        bf1[1] = Wl[(k0 + koff + 1) * MM + a1];
        d0 = __builtin_amdgcn_wmma_f32_16x16x4_f32(false, af, false, bf0, (short)0, d0, false, false);
        d1 = __builtin_amdgcn_wmma_f32_16x16x4_f32(false, af, false, bf1, (short)0, d1, false, false);
    }
    #pragma unroll
    for (int v = 0; v < 8; ++v) {
        int smp = mhalf + v;
        int i0 = (n0 + smp) * MM + a0, i1 = (n0 + smp) * MM + a1;
        float zv0 = zin ? zin[i0] : w_w[a0];
        float zv1 = zin ? zin[i1] : w_w[a1];
        zout[i0] = zv0 + HH * d0[v];
        zout[i1] = zv1 + HH * d1[v];
    }
}

// ----------------------------------------------------------------- grad ----
// out[n][k] = sum_m tanhop*z1*W0[m][k] + sum_j xf[n][j]*symA[j][k] + c_w[k]
__global__ __launch_bounds__(64) void grad_kernel(
    const float* __restrict__ tanhop, const float* __restrict__ z1,
    const float* __restrict__ xf, const float* __restrict__ W0,
    const float* __restrict__ symA, const float* __restrict__ c_w,
    float* __restrict__ out) {
    __shared__ float s[MM];
    __shared__ float xr[DIN];
    int n = blockIdx.x, tid = threadIdx.x;
    for (int i = tid; i < MM; i += 64) s[i] = tanhop[n * MM + i] * z1[n * MM + i];
    xr[tid] = xf[n * DIN + tid];
    __syncthreads();
    float g = c_w[tid];
    for (int m = 0; m < MM; ++m) g += s[m] * W0[m * DIN + tid];
    for (int j = 0; j < DIN; ++j) g += xr[j] * symA[j * DIN + tid];
    out[n * DIN + tid] = g;
}

// --------------------------------------------- fused Jacobian/trH (WMMA) ----
// One workgroup per sample n. J_n (256 x 64, col 63 = zero pad) lives in LDS,
// stored transposed Jt[dd][m] so B-fragments are contiguous float2 loads.
// Two iterations of: KJ = Ws_i @ J (f32 WMMA, K=256), trH += h*sum(wv[m]*KJ^2),
// J += h*tanh(pre_i)∘KJ (in place after barrier; KJ held in VGPR accumulators).
__global__ __launch_bounds__(256) void jac_trh_kernel(
    const float* __restrict__ Ws,       // (2,256,256) row-major
    const float* __restrict__ W0,       // (256,64)
    const float* __restrict__ w_w,      // (256)
    const float* __restrict__ tanhop,
    const float* __restrict__ tp0,
    const float* __restrict__ tp1,
    const float* __restrict__ z2,
    const float* __restrict__ z1,
    const float* __restrict__ rowsum2,
    const float* __restrict__ traceA,
    float* __restrict__ trH) {
    __shared__ float Jt[64 * MM];   // 64 KB: Jt[dd][m]
    __shared__ float tpa[MM];       // tanh(pre0) row
    __shared__ float w1s[MM];       // deriv_tanh(pre0)*z2 row
    __shared__ float w2s[MM];       // deriv_tanh(pre1)*w_w
    __shared__ float sred[MM];

    const int n    = blockIdx.x;
    const int tid  = threadIdx.x;
    const int lane = tid & 31;
    const int wave = tid >> 5;
    const int laneLo = lane & 15;          // M index (A) / N index (B,D)
    const int koff   = (lane >> 4) << 1;   // K offset: lanes 16-31 handle K+2,K+3
    const int mhalf  = (lane >> 4) << 3;   // D rows: lanes 16-31 hold M=v+8

    float to = tanhop[n * MM + tid];
    float t0 = tp0[n * MM + tid];
    float t1 = tp1[n * MM + tid];
    tpa[tid] = t0;
    w1s[tid] = (1.0f - t0 * t0) * z2[n * MM + tid];
    w2s[tid] = (1.0f - t1 * t1) * w_w[tid];
    // trH base: deriv_tanh(opening)*z1*rowsum2 (no h factor)
    float acc = (1.0f - to * to) * z1[n * MM + tid] * rowsum2[tid];

    // J_n[m][dd] = Kopen[m][dd]*tanhopen -> Jt[dd][m]; pad col 63 with 0
    for (int dd = 0; dd < 64; ++dd)
        Jt[dd * MM + tid] = (dd < DD) ? (W0[tid * DIN + dd] * to) : 0.0f;
    __syncthreads();

    const v8f vzero = {0, 0, 0, 0, 0, 0, 0, 0};
    for (int iter = 0; iter < 2; ++iter) {
        const float* Wp = Ws + iter * MM * MM;
        v8f dacc[2][4];
        #pragma unroll
        for (int mi = 0; mi < 2; ++mi)
            #pragma unroll
            for (int nt = 0; nt < 4; ++nt) dacc[mi][nt] = vzero;

        const int m0 = (wave * 2 + 0) * 16 + laneLo;
        const int m1 = (wave * 2 + 1) * 16 + laneLo;
        for (int k0 = 0; k0 < MM; k0 += 4) {
            // A fragments (ISA 32-bit A 16x4 layout): float2 = (K+koff, K+koff+1)
            v2f a0 = *reinterpret_cast<const v2f*>(Wp + m0 * MM + k0 + koff);
            v2f a1 = *reinterpret_cast<const v2f*>(Wp + m1 * MM + k0 + koff);
            // B fragments from LDS (J transposed -> contiguous in K)
            v2f b0 = *reinterpret_cast<const v2f*>(&Jt[(0 * 16 + laneLo) * MM + k0 + koff]);
            v2f b1 = *reinterpret_cast<const v2f*>(&Jt[(1 * 16 + laneLo) * MM + k0 + koff]);
            v2f b2 = *reinterpret_cast<const v2f*>(&Jt[(2 * 16 + laneLo) * MM + k0 + koff]);
            v2f b3 = *reinterpret_cast<const v2f*>(&Jt[(3 * 16 + laneLo) * MM + k0 + koff]);
            dacc[0][0] = __builtin_amdgcn_wmma_f32_16x16x4_f32(false, a0, false, b0, (short)0, dacc[0][0], false, false);
            dacc[0][1] = __builtin_amdgcn_wmma_f32_16x16x4_f32(false, a0, false, b1, (short)0, dacc[0][1], false, false);
            dacc[0][2] = __builtin_amdgcn_wmma_f32_16x16x4_f32(false, a0, false, b2, (short)0, dacc[0][2], false, false);
            dacc[0][3] = __builtin_amdgcn_wmma_f32_16x16x4_f32(false, a0, false, b3, (short)0, dacc[0][3], false, false);
            dacc[1][0] = __builtin_amdgcn_wmma_f32_16x16x4_f32(false, a1, false, b0, (short)0, dacc[1][0], false, false);
            dacc[1][1] = __builtin_amdgcn_wmma_f32_16x16x4_f32(false, a1, false, b1, (short)0, dacc[1][1], false, false);
            dacc[1][2] = __builtin_amdgcn_wmma_f32_16x16x4_f32(false, a1, false, b2, (short)0, dacc[1][2], false, false);
            dacc[1][3] = __builtin_amdgcn_wmma_f32_16x16x4_f32(false, a1, false, b3, (short)0, dacc[1][3], false, false);
        }
        __syncthreads();  // all waves finished reading Jt

        const bool last = (iter == 1);
        const float* wv = last ? w2s : w1s;
        #pragma unroll
        for (int mi = 0; mi < 2; ++mi) {
            const int mbase = (wave * 2 + mi) * 16 + mhalf;  // D layout: M = v (+8 hi lanes)
            #pragma unroll
            for (int nt = 0; nt < 4; ++nt) {
                const int dd = nt * 16 + laneLo;             // D layout: N = lane%16
                #pragma unroll
                for (int v = 0; v < 8; ++v) {
                    float dv = dacc[mi][nt][v];
                    int m = mbase + v;
                    if (dd < DD) {
                        acc += HH * wv[m] * dv * dv;              // trH += h * t_i
                        if (!last) Jt[dd * MM + m] += HH * tpa[m] * dv;  // Jac update
                    }
                }
            }
        }
        __syncthreads();
    }

    // block reduction over the 256 m-threads / tile owners
    sred[tid] = acc;
    __syncthreads();
    for (int off = 128; off > 0; off >>= 1) {
        if (tid < off) sred[tid] += sred[tid + off];
        __syncthreads();
    }
    if (tid == 0) trH[n] = sred[0] + traceA[0];
}

// -------------------------------------------------------------- launcher ----
extern "C" void kernel_launch(void* const* d_in, const int* in_sizes, int n_in,
                              void* d_out, int out_size, void* d_ws, size_t ws_size,
                              hipStream_t stream) {
    const float* t   = (const float*)d_in[0];
    const float* x   = (const float*)d_in[1];
    const float* A   = (const float*)d_in[2];
    const float* W0  = (const float*)d_in[3];
    const float* b0  = (const float*)d_in[4];
    const float* Ws  = (const float*)d_in[5];
    const float* bs  = (const float*)d_in[6];
    const float* w_w = (const float*)d_in[7];
    const float* c_w = (const float*)d_in[8];
    float* out = (float*)d_out;
    float* ws  = (float*)d_ws;

    float* xf      = ws;                    // 4096*64
    float* symA    = xf + NEX * DIN;        // 64*64
    float* rowsum2 = symA + DIN * DIN;      // 256
    float* traceA  = rowsum2 + MM;          // 1 (padded to 16)
    float* tanhop  = traceA + 16;           // 4096*256 each below
    float* u0      = tanhop + NEX * MM;
    float* u1      = u0 + NEX * MM;
    float* tp0     = u1 + NEX * MM;
    float* tp1     = tp0 + NEX * MM;
    float* z2      = tp1 + NEX * MM;
    float* z1      = z2 + NEX * MM;

    setup_kernel<<<1, 256, 0, stream>>>(A, W0, symA, rowsum2, traceA);
    xf_kernel<<<(NEX * DIN + 255) / 256, 256, 0, stream>>>(x, t, xf);
    opening_wmma_kernel<<<NEX / NB, 256, 0, stream>>>(xf, W0, b0, tanhop, u0);
    layer_wmma_kernel<<<NEX / NB, 256, 0, stream>>>(u0, Ws,           bs,      tp0, u1);
    layer_wmma_kernel<<<NEX / NB, 256, 0, stream>>>(u1, Ws + MM * MM, bs + MM, tp1, nullptr);
    zback_wmma_kernel<<<NEX / NB, 256, 0, stream>>>(tp1, nullptr, w_w, Ws + MM * MM, z2);
    zback_wmma_kernel<<<NEX / NB, 256, 0, stream>>>(tp0, z2,      w_w, Ws,           z1);
    grad_kernel<<<NEX, 64, 0, stream>>>(tanhop, z1, xf, W0, symA, c_w, out);
    jac_trh_kernel<<<NEX, 256, 0, stream>>>(Ws, W0, w_w, tanhop, tp0, tp1, z2, z1,
                                            rowsum2, traceA, out + NEX * DIN);
}